// SpatialGather_Module_2894807957610
// MI455X (gfx1250) — compile-verified
//
#include <hip/hip_runtime.h>

typedef __attribute__((ext_vector_type(2))) float v2f;
typedef __attribute__((ext_vector_type(8))) float v8f;

#define BB 8
#define KK 19
#define CC 512
#define SS 32768
#define KPAD 32
#define SCHUNKS 16
#define SCHUNK (SS / SCHUNKS)   /* 2048 */
#define CTILE 64
#define STILE 64
#define LSTR 68                 /* LDS row stride in floats: 272B, 16B-aligned, bank-conflict-free */

// ---------------- Pass 1: row softmax over the spatial dim ----------------
__global__ __launch_bounds__(256) void softmax_rows(const float* __restrict__ probs,
                                                    float* __restrict__ wbuf) {
  const int row = blockIdx.x;                 // 0 .. B*K-1
  const float* p = probs + (size_t)row * SS;
  float* w = wbuf + (size_t)row * SS;
  __shared__ float red[256];
  const int tid = threadIdx.x;

  float m = -__builtin_inff();
  for (int s = tid; s < SS; s += 256) m = fmaxf(m, p[s]);
  red[tid] = m; __syncthreads();
  for (int off = 128; off > 0; off >>= 1) {
    if (tid < off) red[tid] = fmaxf(red[tid], red[tid + off]);
    __syncthreads();
  }
  const float rmax = red[0]; __syncthreads();

  float acc = 0.0f;
  for (int s = tid; s < SS; s += 256) acc += __expf(p[s] - rmax);
  red[tid] = acc; __syncthreads();
  for (int off = 128; off > 0; off >>= 1) {
    if (tid < off) red[tid] += red[tid + off];
    __syncthreads();
  }
  const float rinv = 1.0f / red[0];

  for (int s = tid; s < SS; s += 256) w[s] = __expf(p[s] - rmax) * rinv;
}

// ---------------- Pass 2: ctx[k,c] = sum_s w[k,s] * f[c,s] via V_WMMA_F32_16X16X4_F32 ----
// grid: x = c-tile (8), y = batch (8), z = s-chunk (16); block = 128 threads (4 waves)
// Each wave: D tile = 16 k-rows x 16 c-cols, two M-tiles (k 0-15, 16-31; K=19 zero-padded),
// each M-tile accumulated in two independent chains (even/odd k-step pairs) for XDL ILP.
__global__ __launch_bounds__(128) void gemm_wmma(const float* __restrict__ feats,
                                                 const float* __restrict__ wbuf,
                                                 float* __restrict__ part) {
  const int ct = blockIdx.x, b = blockIdx.y, sc = blockIdx.z;
  const int c0 = ct * CTILE;
  const int s_begin = sc * SCHUNK;
  const float* f = feats + (size_t)b * CC * SS;
  const float* w = wbuf + (size_t)b * KK * SS;

  __shared__ float ftile[CTILE][LSTR];   // [c][s] padded
  __shared__ float wtile[KPAD][LSTR];    // [k][s] padded, rows >= 19 zero

  const int tid = threadIdx.x;           // 0..127
  const int lane = tid & 31;
  const int wv = tid >> 5;               // wave id 0..3 -> c sub-tile
  const int hi = lane >> 4;              // 0: K pair {0,1}; 1: K pair {2,3}
  const int lo = lane & 15;

  v8f acc0a = {}, acc0b = {};            // k rows 0..15 (two chains)
  v8f acc1a = {}, acc1b = {};            // k rows 16..31 (two chains; only 16..18 valid)

  for (int st = 0; st < SCHUNK; st += STILE) {
    const int sbase = s_begin + st;
    __syncthreads();
    // Stage feats tile: 64 rows x 64 floats = 1024 float4 (coalesced, read-once stream)
#pragma unroll
    for (int i = 0; i < 8; ++i) {
      const int li = i * 128 + tid;      // 0..1023
      const int r = li >> 4;             // c row 0..63
      const int q = li & 15;             // float4 column
      const float4 v = *reinterpret_cast<const float4*>(
          f + (size_t)(c0 + r) * SS + sbase + q * 4);
      *reinterpret_cast<float4*>(&ftile[r][q * 4]) = v;
    }
    // Stage weight tile: 32 rows x 64 floats = 512 float4; rows >= 19 zeroed
#pragma unroll
    for (int i = 0; i < 4; ++i) {
      const int li = i * 128 + tid;      // 0..511
      const int r = li >> 4;             // k row 0..31
      const int q = li & 15;
      float4 v = make_float4(0.0f, 0.0f, 0.0f, 0.0f);
      if (r < KK)
        v = *reinterpret_cast<const float4*>(w + (size_t)r * SS + sbase + q * 4);
      *reinterpret_cast<float4*>(&wtile[r][q * 4]) = v;
    }
    __syncthreads();
    // 8 outer steps of 8 s-values: two k=4 sub-steps feeding independent accumulator chains.
#pragma unroll
    for (int ks = 0; ks < STILE; ks += 8) {
      const int sA = ks + 2 * hi;        // lanes 0-15: K={0,1}; lanes 16-31: K={2,3}
      const int sB = ks + 4 + 2 * hi;
      // B fragments: 4(s) x 16(c); lane col = c, VGPR0/1 = consecutive s -> b64 LDS reads
      const v2f bfragA = *reinterpret_cast<const v2f*>(&ftile[wv * 16 + lo][sA]);
      const v2f bfragB = *reinterpret_cast<const v2f*>(&ftile[wv * 16 + lo][sB]);
      // A fragments: 16(k) x 4(s); lane row = k
      const v2f a0A = *reinterpret_cast<const v2f*>(&wtile[lo][sA]);
      const v2f a1A = *reinterpret_cast<const v2f*>(&wtile[16 + lo][sA]);
      const v2f a0B = *reinterpret_cast<const v2f*>(&wtile[lo][sB]);
      const v2f a1B = *reinterpret_cast<const v2f*>(&wtile[16 + lo][sB]);
      acc0a = __builtin_amdgcn_wmma_f32_16x16x4_f32(false, a0A, false, bfragA,
                                                    (short)0, acc0a, false, false);
      acc1a = __builtin_amdgcn_wmma_f32_16x16x4_f32(false, a1A, false, bfragA,
                                                    (short)0, acc1a, false, false);
      acc0b = __builtin_amdgcn_wmma_f32_16x16x4_f32(false, a0B, false, bfragB,
                                                    (short)0, acc0b, false, false);
      acc1b = __builtin_amdgcn_wmma_f32_16x16x4_f32(false, a1B, false, bfragB,
                                                    (short)0, acc1b, false, false);
    }
  }

  // Merge chains and store partials: part[((sc*B + b)*C + c)*KPAD + k]
  // D layout: VGPR r at lane (hi,lo) -> k = r + 8*hi, c-col = lo.
  const int ccol = c0 + wv * 16 + lo;
  float* pbase = part + (((size_t)sc * BB + b) * CC + ccol) * KPAD;
#pragma unroll
  for (int r = 0; r < 8; ++r) {
    pbase[r + 8 * hi] = acc0a[r] + acc0b[r];
    pbase[16 + r + 8 * hi] = acc1a[r] + acc1b[r];
  }
}

// ---------------- Pass 3: deterministic reduction of s-chunk partials ----------------
__global__ __launch_bounds__(256) void reduce_out(const float* __restrict__ part,
                                                  float* __restrict__ out) {
  const int idx = blockIdx.x * blockDim.x + threadIdx.x;  // over B*C*K
  if (idx >= BB * CC * KK) return;
  const int k = idx % KK;
  const int c = (idx / KK) % CC;
  const int b = idx / (KK * CC);
  float s = 0.0f;
#pragma unroll
  for (int sc = 0; sc < SCHUNKS; ++sc)
    s += part[(((size_t)sc * BB + b) * CC + c) * KPAD + k];
  out[idx] = s;   // out flat = ((b*C + c)*K + k), matching [B, C, K, 1]
}

extern "C" void kernel_launch(void* const* d_in, const int* in_sizes, int n_in,
                              void* d_out, int out_size, void* d_ws, size_t ws_size,
                              hipStream_t stream) {
  const float* feats = (const float*)d_in[0];   // [8, 512, 128, 256] f32
  const float* probs = (const float*)d_in[1];   // [8, 19, 128, 256] f32
  float* out = (float*)d_out;                   // [8, 512, 19, 1] f32
  float* wbuf = (float*)d_ws;                              // B*K*S floats (~19.9 MB)
  float* part = wbuf + (size_t)BB * KK * SS;               // SCHUNKS*B*C*KPAD floats (~8 MB)

  softmax_rows<<<BB * KK, 256, 0, stream>>>(probs, wbuf);

  dim3 grid(CC / CTILE, BB, SCHUNKS);           // 8 x 8 x 16 = 1024 blocks
  gemm_wmma<<<grid, 128, 0, stream>>>(feats, wbuf, part);

  const int tot = BB * CC * KK;
  reduce_out<<<(tot + 255) / 256, 256, 0, stream>>>(part, out);
}